// MinibatchDiscrimination_74010876444714
// MI455X (gfx1250) — compile-verified
//
#include <hip/hip_runtime.h>

typedef __attribute__((ext_vector_type(2))) float v2f;
typedef __attribute__((ext_vector_type(8))) float v8f;

// -----------------------------------------------------------------------------
// Kernel 1: m = x @ W^T + b  via V_WMMA_F32_16X16X4_F32 (full fp32 precision)
// x: 512 x 1024, W: 1024 x 1024 (row n of W is column n of W^T), m: 512 x 1024
// One wave per 16x16 output tile. 2048 tiles total; 8 waves per 256-thr block.
// A-operand layout (fp32 16x4): lanes 0-15 -> {K=0,K=1}, lanes 16-31 -> {K=2,K=3}
// -----------------------------------------------------------------------------
__global__ __launch_bounds__(256) void mbd_gemm_wmma_f32(
    const float* __restrict__ X, const float* __restrict__ W,
    const float* __restrict__ bias, float* __restrict__ M)
{
    const int wave = threadIdx.x >> 5;
    const int tile = blockIdx.x * 8 + wave;   // 0..2047
    const int tm   = tile >> 6;               // 0..31  (row tile, 512/16)
    const int tn   = tile & 63;               // 0..63  (col tile, 1024/16)
    const int lane = threadIdx.x & 31;
    const int lr   = lane & 15;
    const int half = lane >> 4;               // 0: K pair {0,1}, 1: K pair {2,3}

    const float* __restrict__ arow = X + (size_t)(tm * 16 + lr) * 1024 + 2 * half;
    const float* __restrict__ brow = W + (size_t)(tn * 16 + lr) * 1024 + 2 * half;

    v8f c = {};
#pragma unroll 8
    for (int k = 0; k < 1024; k += 4) {
        v2f a = *(const v2f*)(arow + k);
        v2f b = *(const v2f*)(brow + k);
        c = __builtin_amdgcn_wmma_f32_16x16x4_f32(
                /*neg_a=*/false, a, /*neg_b=*/false, b,
                /*c_mod=*/(short)0, c, /*reuse_a=*/false, /*reuse_b=*/false);
    }

    const float bb = bias[tn * 16 + lr];
    float* __restrict__ out = M + (size_t)(tm * 16 + half * 8) * 1024 + tn * 16 + lr;
#pragma unroll
    for (int r = 0; r < 8; ++r)
        out[(size_t)r * 1024] = c[r] + bb;    // VGPR r -> rows M=r (lanes 0-15), M=r+8 (lanes 16-31)
}

// -----------------------------------------------------------------------------
// Kernel 2: o[j,B] = sum_i exp(-sum_c |m[i,B,c]-m[j,B,c]|)   (512x64 outputs)
// One thread per (j,B). Lane index == B -> per-i loads are coalesced (64B/lane).
// Writes into columns 1024..1087 of the 1088-wide output.
// -----------------------------------------------------------------------------
__global__ __launch_bounds__(256) void mbd_pairwise_exp_l1(
    const float* __restrict__ M, float* __restrict__ Out)
{
    const int gid = blockIdx.x * 256 + threadIdx.x;   // 0..32767
    const int j = gid >> 6;
    const int B = gid & 63;

    const float4* __restrict__ mj = (const float4*)(M + (size_t)j * 1024 + B * 16);
    const float4 r0 = mj[0], r1 = mj[1], r2 = mj[2], r3 = mj[3];

    float acc = 0.0f;
#pragma unroll 2
    for (int i = 0; i < 512; ++i) {
        const float4* __restrict__ mi = (const float4*)(M + (size_t)i * 1024 + B * 16);
        const float4 a0 = mi[0], a1 = mi[1], a2 = mi[2], a3 = mi[3];
        float d = fabsf(a0.x - r0.x) + fabsf(a0.y - r0.y) + fabsf(a0.z - r0.z) + fabsf(a0.w - r0.w)
                + fabsf(a1.x - r1.x) + fabsf(a1.y - r1.y) + fabsf(a1.z - r1.z) + fabsf(a1.w - r1.w)
                + fabsf(a2.x - r2.x) + fabsf(a2.y - r2.y) + fabsf(a2.z - r2.z) + fabsf(a2.w - r2.w)
                + fabsf(a3.x - r3.x) + fabsf(a3.y - r3.y) + fabsf(a3.z - r3.z) + fabsf(a3.w - r3.w);
        acc += __expf(-d);
    }
    Out[(size_t)j * 1088 + 1024 + B] = acc;
}

// -----------------------------------------------------------------------------
// Kernel 3: copy x into columns 0..1023 of the output (float4, both 16B aligned)
// -----------------------------------------------------------------------------
__global__ __launch_bounds__(256) void mbd_copy_x(
    const float* __restrict__ X, float* __restrict__ Out)
{
    const int idx = blockIdx.x * 256 + threadIdx.x;   // 0..131071 (512*256 float4)
    const int j = idx >> 8;                           // row
    const int c = idx & 255;                          // float4 column
    ((float4*)(Out + (size_t)j * 1088))[c] = ((const float4*)(X + (size_t)j * 1024))[c];
}

extern "C" void kernel_launch(void* const* d_in, const int* in_sizes, int n_in,
                              void* d_out, int out_size, void* d_ws, size_t ws_size,
                              hipStream_t stream) {
    const float* x = (const float*)d_in[0];   // 512*1024
    const float* W = (const float*)d_in[1];   // 1024*1024
    const float* b = (const float*)d_in[2];   // 1024
    float* out = (float*)d_out;               // 512*1088
    float* m   = (float*)d_ws;                // 512*1024 fp32 scratch (2 MB)

    // 2048 tiles / 8 waves per block = 256 blocks
    mbd_gemm_wmma_f32<<<256, 256, 0, stream>>>(x, W, b, m);
    // 512*256 float4 copies / 256 = 512 blocks (independent of kernel 1)
    mbd_copy_x<<<512, 256, 0, stream>>>(x, out);
    // 32768 (j,B) threads / 256 = 128 blocks (stream-ordered after the GEMM)
    mbd_pairwise_exp_l1<<<128, 256, 0, stream>>>(m, out);
}